// Net_19335942766759
// MI455X (gfx1250) — compile-verified
//
#include <hip/hip_runtime.h>
#include <hip/hip_bf16.h>
#include <math.h>

// Problem constants (match reference)
#define NNODE 100000
#define NEDGE 800000
#define NMOTIF 13
#define K_IN 500
#define HID 128
#define CD 6
#define NC 3

typedef float v2f __attribute__((ext_vector_type(2)));
typedef float v8f __attribute__((ext_vector_type(8)));
typedef unsigned int u32x4 __attribute__((ext_vector_type(4)));
typedef int i32x4 __attribute__((ext_vector_type(4)));
typedef int i32x8 __attribute__((ext_vector_type(8)));

// ---------------------------------------------------------------------------
// K1: h0 = h @ W0 + b0 via V_WMMA_F32_16X16X4_F32 (fp32, full precision).
// grid = N/16 blocks, 256 threads = 8 waves; wave w computes the 16x16 tile at
// rows blockIdx.x*16, cols w*16.
//
// A-tile (16 x 500 fp32, 32 KB) is DMA'd once per block into LDS by the
// Tensor Data Mover (tensor_load_to_lds, TENSORcnt), then all 8 waves read
// their A fragments with ds_load_b64. Bank math: addr dwords = 500*row + k;
// 500 mod 64 = 52, gcd(52,64)=4 -> the 16 rows map to 16 distinct banks
// (multiples of 4), the upper half-wave reads k+2 (offset 2 banks), so the
// 32 b64 reads cover all 64 banks conflict-free without padding.
//
// WMMA VGPR layouts per CDNA5 ISA 7.12.2:
//   A 16x4 f32: lanes 0-15 -> M=lane, {K0,K1}; lanes 16-31 -> M=lane-16, {K2,K3}
//   B 4x16 f32: lanes 0-15 -> N=lane, rows {K0,K1}; lanes 16-31 -> rows {K2,K3}
//   C/D: VGPR j: lanes 0-15 -> M=j, N=lane; lanes 16-31 -> M=8+j, N=lane-16
// ---------------------------------------------------------------------------
__global__ __launch_bounds__(256) void gemm_wmma_f32(
    const float* __restrict__ A, const float* __restrict__ B,
    const float* __restrict__ bias, float* __restrict__ C) {
  __shared__ float tileA[16 * K_IN];  // 32000 B

  const int lane = threadIdx.x & 31;
  const int wv   = threadIdx.x >> 5;     // column tile 0..7
  const int row0 = blockIdx.x * 16;
  const int col0 = wv * 16;
  const int half = lane >> 4;            // 0: lanes 0-15, 1: lanes 16-31
  const int l16  = lane & 15;

  // --- TDM: stage the 16x500 A tile into LDS (wave 0 issues; EXEC ignored) ---
  if (threadIdx.x < 32) {
    const unsigned ldsAddr =
        (unsigned)(unsigned long long)(uintptr_t)(void*)tileA;  // LDS byte offset
    const unsigned long long gaddr =
        (unsigned long long)(uintptr_t)(A + (size_t)row0 * K_IN);

    u32x4 g0;
    g0[0] = 1u;                                   // count=1 (valid), user mode
    g0[1] = ldsAddr;                              // lds_addr[31:0]
    g0[2] = (unsigned)(gaddr & 0xFFFFFFFFu);      // global_addr[31:0]
    g0[3] = (unsigned)((gaddr >> 32) & 0x1FFFFFFu)  // global_addr[56:32]
            | (2u << 30);                         // type=2 ("image")

    i32x8 g1;
    g1[0] = (int)(2u << 16);      // workgroup_mask=0, data_size=2 (4 bytes)
    g1[1] = (int)(500u << 16);    // atomic_barrier_addr=0, tensor_dim0[15:0]=500
    g1[2] = (int)(16u << 16);     // tensor_dim0[31:16]=0, tensor_dim1[15:0]=16
    g1[3] = (int)(500u << 16);    // tensor_dim1[31:16]=0, tile_dim0=500
    g1[4] = 16;                   // tile_dim1=16, tile_dim2=0
    g1[5] = 500;                  // tensor_dim0_stride[31:0]=500
    g1[6] = 0;                    // stride hi / tensor_dim1_stride lo = 0
    g1[7] = 0;                    // tensor_dim1_stride hi = 0

    i32x4 gz = {0, 0, 0, 0};      // groups 2/3 unused (2D tile)
#if __clang_major__ >= 23
    i32x8 gz8 = {0, 0, 0, 0, 0, 0, 0, 0};
    __builtin_amdgcn_tensor_load_to_lds(g0, g1, gz, gz, gz8, 0);
#else
    __builtin_amdgcn_tensor_load_to_lds(g0, g1, gz, gz, 0);
#endif
    __builtin_amdgcn_s_wait_tensorcnt(0);
  }
  __syncthreads();

  // --- WMMA main loop: A from LDS (ds_load_b64), B from L2-resident W0 ---
  v8f acc;
  const float bv = bias[col0 + l16];
#pragma unroll
  for (int j = 0; j < 8; ++j) acc[j] = bv;

  const float* arow = tileA + l16 * K_IN;   // this lane's A row in LDS
  for (int k0 = 0; k0 < K_IN; k0 += 4) {
    v2f af = *(const v2f*)(arow + k0 + 2 * half);   // 8B-aligned ds_load_b64
    v2f bf;
    const float* bp = B + (size_t)(k0 + 2 * half) * HID + col0 + l16;
    bf[0] = bp[0];
    bf[1] = bp[HID];
    acc = __builtin_amdgcn_wmma_f32_16x16x4_f32(
        false, af, false, bf, (short)0, acc, false, false);
  }

#pragma unroll
  for (int j = 0; j < 8; ++j) {
    const int r = row0 + j + 8 * half;
    C[(size_t)r * HID + col0 + l16] = acc[j];
  }
}

// ---------------------------------------------------------------------------
// Zero fill (workspace m buffer) — grid-stride
// ---------------------------------------------------------------------------
__global__ void zero_f32(float* __restrict__ p, int n) {
  int i = blockIdx.x * blockDim.x + threadIdx.x;
  const int stride = gridDim.x * blockDim.x;
  for (; i < n; i += stride) p[i] = 0.0f;
}

// ---------------------------------------------------------------------------
// K2: edge scatter-add  m[dst] += h0[src] * w   (one wave per edge,
// each lane handles a float4 chunk of the 128-wide feature row).
// h0 (51.2 MB) is L2-resident; atomics are hardware global_atomic_add_f32.
// ---------------------------------------------------------------------------
__global__ __launch_bounds__(256) void scatter_edges(
    const float* __restrict__ h0, const int* __restrict__ src,
    const int* __restrict__ dst, const float* __restrict__ ew,
    float* __restrict__ m, int E) {
  const int e = blockIdx.x * (blockDim.x >> 5) + (threadIdx.x >> 5);
  const int lane = threadIdx.x & 31;
  if (e >= E) return;
  const int s = src[e];
  const int d = dst[e];
  const float w = ew[e];
  const float4 v = ((const float4*)(h0 + (size_t)s * HID))[lane];
  float* md = m + (size_t)d * HID + lane * 4;
  unsafeAtomicAdd(md + 0, v.x * w);
  unsafeAtomicAdd(md + 1, v.y * w);
  unsafeAtomicAdd(md + 2, v.z * w);
  unsafeAtomicAdd(md + 3, v.w * w);
}

// ---------------------------------------------------------------------------
// K3: per-node compress + attention for one motif:
//   c = [h0 ; m] @ Wc + bc ;  a = sigmoid(c . att_w) ;  hc[:,col..col+5] = a*c
// ---------------------------------------------------------------------------
__global__ __launch_bounds__(128) void compress_att(
    const float* __restrict__ h0, const float* __restrict__ m,
    const float* __restrict__ Wc, const float* __restrict__ bc,
    const float* __restrict__ aw, float* __restrict__ hc, int colOff) {
  const int n = blockIdx.x * blockDim.x + threadIdx.x;
  if (n >= NNODE) return;

  float acc[CD];
#pragma unroll
  for (int j = 0; j < CD; ++j) acc[j] = bc[j];

  const float* x0 = h0 + (size_t)n * HID;
  for (int k = 0; k < HID; ++k) {
    const float x = x0[k];
    const float* w = Wc + (size_t)k * CD;
#pragma unroll
    for (int j = 0; j < CD; ++j) acc[j] += x * w[j];
  }
  const float* x1 = m + (size_t)n * HID;
  for (int k = 0; k < HID; ++k) {
    const float x = x1[k];
    const float* w = Wc + (size_t)(HID + k) * CD;
#pragma unroll
    for (int j = 0; j < CD; ++j) acc[j] += x * w[j];
  }

  float s = 0.0f;
#pragma unroll
  for (int j = 0; j < CD; ++j) s += acc[j] * aw[j];
  const float a = 1.0f / (1.0f + __expf(-s));

  float* o = hc + (size_t)n * (NMOTIF * CD) + colOff;
#pragma unroll
  for (int j = 0; j < CD; ++j) o[j] = a * acc[j];
}

// ---------------------------------------------------------------------------
// K4: classifier head  out = relu(hc) @ Wd + bd   ([N,78] x [78,3])
// ---------------------------------------------------------------------------
__global__ __launch_bounds__(256) void head_dense(
    const float* __restrict__ hc, const float* __restrict__ Wd,
    const float* __restrict__ bd, float* __restrict__ out) {
  const int n = blockIdx.x * blockDim.x + threadIdx.x;
  if (n >= NNODE) return;
  float a0 = bd[0], a1 = bd[1], a2 = bd[2];
  const float* r = hc + (size_t)n * (NMOTIF * CD);
#pragma unroll 6
  for (int k = 0; k < NMOTIF * CD; ++k) {
    const float x = fmaxf(r[k], 0.0f);
    a0 += x * Wd[k * NC + 0];
    a1 += x * Wd[k * NC + 1];
    a2 += x * Wd[k * NC + 2];
  }
  out[(size_t)n * NC + 0] = a0;
  out[(size_t)n * NC + 1] = a1;
  out[(size_t)n * NC + 2] = a2;
}

// ---------------------------------------------------------------------------
extern "C" void kernel_launch(void* const* d_in, const int* in_sizes, int n_in,
                              void* d_out, int out_size, void* d_ws, size_t ws_size,
                              hipStream_t stream) {
  const float* h   = (const float*)d_in[0];   // [N, IN]
  const int*   src = (const int*)  d_in[1];   // [M, E]
  const int*   dst = (const int*)  d_in[2];   // [M, E]
  const float* ew  = (const float*)d_in[3];   // [M, E]
  const float* W0  = (const float*)d_in[4];   // [IN, HID]
  const float* b0  = (const float*)d_in[5];   // [HID]
  const float* Wc  = (const float*)d_in[6];   // [M, 2*HID, CD]
  const float* bc  = (const float*)d_in[7];   // [M, CD]
  const float* aw  = (const float*)d_in[8];   // [M, CD]
  const float* Wd  = (const float*)d_in[9];   // [M*CD, NC]
  const float* bd  = (const float*)d_in[10];  // [NC]
  float* out = (float*)d_out;

  // workspace layout
  float* h0 = (float*)d_ws;                       // N*HID
  float* m  = h0 + (size_t)NNODE * HID;           // N*HID (reused per motif)
  float* hc = m  + (size_t)NNODE * HID;           // N * (M*CD)

  // K1: TDM-staged WMMA GEMM h0 = h @ W0 + b0
  gemm_wmma_f32<<<NNODE / 16, 256, 0, stream>>>(h, W0, b0, h0);

  // Per-motif: zero m, scatter edges, compress + attention
  const int edgeBlocks = (NEDGE + 7) / 8;  // 8 edges (waves) per 256-thread block
  for (int i = 0; i < NMOTIF; ++i) {
    zero_f32<<<4096, 256, 0, stream>>>(m, NNODE * HID);
    scatter_edges<<<edgeBlocks, 256, 0, stream>>>(
        h0, src + (size_t)i * NEDGE, dst + (size_t)i * NEDGE,
        ew + (size_t)i * NEDGE, m, NEDGE);
    compress_att<<<(NNODE + 127) / 128, 128, 0, stream>>>(
        h0, m, Wc + (size_t)i * 2 * HID * CD, bc + (size_t)i * CD,
        aw + (size_t)i * CD, hc, i * CD);
  }

  // K4: head
  head_dense<<<(NNODE + 255) / 256, 256, 0, stream>>>(hc, Wd, bd, out);
}